// SGC_CUSTOM_7722351198606
// MI455X (gfx1250) — compile-verified
//
#include <hip/hip_runtime.h>

#define N_NODES 100000
#define D_FEAT 64
#define N_CLASSES 40
#define WT_STRIDE 68   // LDS row stride (dwords): 68 mod 64 = 4 -> conflict-free b64 reads

typedef __attribute__((ext_vector_type(2))) float v2f;
typedef __attribute__((ext_vector_type(8))) float v8f;

// ---------------- degree / norm ----------------

__global__ __launch_bounds__(256) void k_fill1(float* p, int n) {
    int i = blockIdx.x * 256 + threadIdx.x;
    if (i < n) p[i] = 1.0f;   // self-loop contributes 1 to every degree
}

__global__ __launch_bounds__(256) void k_deg(float* __restrict__ deg,
                                             const int* __restrict__ col, int e) {
    int i = blockIdx.x * 256 + threadIdx.x;
    if (i < e) unsafeAtomicAdd(deg + col[i], 1.0f);
}

__global__ __launch_bounds__(256) void k_rsqrt_inplace(float* p, int n) {
    int i = blockIdx.x * 256 + threadIdx.x;
    if (i < n) p[i] = rsqrtf(p[i]);   // deg >= 1 always (self loops)
}

__global__ __launch_bounds__(256) void k_norm(float* __restrict__ norm,
                                              const float* __restrict__ dinv,
                                              const int* __restrict__ row,
                                              const int* __restrict__ col, int e) {
    int i = blockIdx.x * 256 + threadIdx.x;
    if (i < e) norm[i] = dinv[row[i]] * dinv[col[i]];
}

// ---------------- one propagation hop ----------------
// out[i,:] starts as dinv[i]^2 * in[i,:]  (the self-loop edge), then scatter.

__global__ __launch_bounds__(256) void k_hop_init(float* __restrict__ out,
                                                  const float* __restrict__ in,
                                                  const float* __restrict__ dinv, int total) {
    int i = blockIdx.x * 256 + threadIdx.x;
    if (i < total) {
        float d = dinv[i >> 6];   // i / D_FEAT
        out[i] = d * d * in[i];
    }
}

// one wave32 per edge; lane handles 2 consecutive features (64 total)
__global__ __launch_bounds__(256) void k_hop_scatter(const float* __restrict__ in,
                                                     float* __restrict__ out,
                                                     const int* __restrict__ row,
                                                     const int* __restrict__ col,
                                                     const float* __restrict__ norm, int e) {
    int wid = (blockIdx.x * 256 + threadIdx.x) >> 5;
    int lane = threadIdx.x & 31;
    if (wid >= e) return;
    int r = row[wid];
    int c = col[wid];
    float w = norm[wid];
    int f = lane * 2;
    const float2 v = *(const float2*)(in + r * 64 + f);
    float* o = out + c * 64 + f;
    unsafeAtomicAdd(o,     w * v.x);
    unsafeAtomicAdd(o + 1, w * v.y);
}

// ---------------- fp32 WMMA GEMM: out[N x outC] = in[N x 64] @ W[64 x outC] + bias ----------------
// W is staged transposed + zero-padded into LDS once per block:
//   Wt[c*WT_STRIDE + k] = W[k*outC + c]  (c < outC), else 0
// One wave per 16x16 output tile. K=64 -> 16 x V_WMMA_F32_16X16X4_F32.
// A 16x4 layout: lane L holds M=L%16, K = 4k + 2*(L/16) + {0,1}
// B 4x16 layout: VGPR j holds N=L%16, K = 4k + 2*(L/16) + j
// C/D 16x16:     VGPR r holds N=L%16, M = r + 8*(L/16)

__global__ __launch_bounds__(256) void k_gemm_wmma(const float* __restrict__ in,
                                                   const float* __restrict__ W,
                                                   const float* __restrict__ bias,
                                                   float* __restrict__ out,
                                                   int nRowTiles, int colTiles, int outC, int relu) {
    __shared__ float Wt[64 * WT_STRIDE];
    __shared__ float bs[64];

    const int padC = colTiles * 16;           // 64 or 48

    // stage transposed, zero-padded W and bias
    for (int idx = threadIdx.x; idx < padC * 64; idx += 256) {
        int c = idx >> 6;         // padded output column
        int k = idx & 63;         // K index
        Wt[c * WT_STRIDE + k] = (c < outC) ? W[k * outC + c] : 0.0f;
    }
    for (int idx = threadIdx.x; idx < padC; idx += 256)
        bs[idx] = (idx < outC) ? bias[idx] : 0.0f;
    __syncthreads();

    int waveId = threadIdx.x >> 5;
    int lane = threadIdx.x & 31;
    int tile = blockIdx.x * 8 + waveId;
    if (tile >= nRowTiles * colTiles) return;   // wave-uniform exit: EXEC full for WMMA

    int rowTile = tile / colTiles;
    int colTile = tile - rowTile * colTiles;
    int m = lane & 15;
    int half = lane >> 4;
    int row0 = rowTile * 16;
    int col0 = colTile * 16;
    int col = col0 + m;

    const float* arow = in + (row0 + m) * 64;            // this lane's A row
    const float* brow = Wt + col * WT_STRIDE + half * 2; // this lane's B column (in LDS)
    v8f acc = {};
#pragma unroll
    for (int k = 0; k < 16; ++k) {
        v2f a, b;
        a = *(const v2f*)(arow + k * 4 + half * 2);
        b = *(const v2f*)(brow + k * 4);
        acc = __builtin_amdgcn_wmma_f32_16x16x4_f32(
            /*neg_a=*/false, a, /*neg_b=*/false, b,
            /*c_mod=*/(short)0, acc, /*reuse_a=*/false, /*reuse_b=*/false);
    }

    float bv = bs[col];
#pragma unroll
    for (int r = 0; r < 8; ++r) {
        int rowi = row0 + r + half * 8;
        float v = acc[r] + bv;
        if (relu) v = fmaxf(v, 0.0f);
        if (col < outC) out[rowi * outC + col] = v;
    }
}

// ---------------- row-wise log_softmax over 40 classes, in place ----------------

__global__ __launch_bounds__(256) void k_logsoftmax(float* out, int n) {
    int i = blockIdx.x * 256 + threadIdx.x;
    if (i >= n) return;
    float* p = out + i * N_CLASSES;
    float vals[N_CLASSES];
    float mx = -3.4e38f;
#pragma unroll
    for (int j = 0; j < N_CLASSES; ++j) { vals[j] = p[j]; mx = fmaxf(mx, vals[j]); }
    float s = 0.0f;
#pragma unroll
    for (int j = 0; j < N_CLASSES; ++j) s += expf(vals[j] - mx);
    float ls = logf(s);
#pragma unroll
    for (int j = 0; j < N_CLASSES; ++j) p[j] = vals[j] - mx - ls;
}

// ---------------- host side ----------------

static inline void launch_hop(const float* in, float* out, const float* dinv,
                              const float* norm, const int* rowp, const int* colp,
                              int e, hipStream_t stream) {
    k_hop_init<<<(N_NODES * D_FEAT + 255) / 256, 256, 0, stream>>>(out, in, dinv, N_NODES * D_FEAT);
    k_hop_scatter<<<(e + 7) / 8, 256, 0, stream>>>(in, out, rowp, colp, norm, e);
}

extern "C" void kernel_launch(void* const* d_in, const int* in_sizes, int n_in,
                              void* d_out, int out_size, void* d_ws, size_t ws_size,
                              hipStream_t stream) {
    const float* x  = (const float*)d_in[0];
    const int*   ei = (const int*)d_in[1];
    const float* W1 = (const float*)d_in[2];
    const float* b1 = (const float*)d_in[3];
    const float* W2 = (const float*)d_in[4];
    const float* b2 = (const float*)d_in[5];
    const float* Wl = (const float*)d_in[6];
    const float* bl = (const float*)d_in[7];
    float* out = (float*)d_out;

    const int E = in_sizes[1] / 2;
    const int* rowp = ei;
    const int* colp = ei + E;

    float* dinv = (float*)d_ws;
    float* norm = dinv + N_NODES;
    float* t0 = norm + E;
    float* t1 = t0 + (size_t)N_NODES * D_FEAT;

    // GCN normalization
    k_fill1<<<(N_NODES + 255) / 256, 256, 0, stream>>>(dinv, N_NODES);
    k_deg<<<(E + 255) / 256, 256, 0, stream>>>(dinv, colp, E);
    k_rsqrt_inplace<<<(N_NODES + 255) / 256, 256, 0, stream>>>(dinv, N_NODES);
    k_norm<<<(E + 255) / 256, 256, 0, stream>>>(norm, dinv, rowp, colp, E);

    const int rt = N_NODES / 16;  // 6250 row tiles

    // SGConv1: P^2 x, then linear+ReLU
    launch_hop(x,  t0, dinv, norm, rowp, colp, E, stream);
    launch_hop(t0, t1, dinv, norm, rowp, colp, E, stream);
    k_gemm_wmma<<<(rt * 4 + 7) / 8, 256, 0, stream>>>(t1, W1, b1, t0, rt, 4, 64, 1);

    // SGConv2
    launch_hop(t0, t1, dinv, norm, rowp, colp, E, stream);
    launch_hop(t1, t0, dinv, norm, rowp, colp, E, stream);
    k_gemm_wmma<<<(rt * 4 + 7) / 8, 256, 0, stream>>>(t0, W2, b2, t1, rt, 4, 64, 1);

    // classifier head: logits -> log_softmax (in place on d_out)
    launch_hop(t1, t0, dinv, norm, rowp, colp, E, stream);
    launch_hop(t0, t1, dinv, norm, rowp, colp, E, stream);
    k_gemm_wmma<<<(rt * 3 + 7) / 8, 256, 0, stream>>>(t1, Wl, bl, out, rt, 3, 40, 0);
    k_logsoftmax<<<(N_NODES + 255) / 256, 256, 0, stream>>>(out, N_NODES);
}